// LNCCLoss_14525579395713
// MI455X (gfx1250) — compile-verified
//
#include <hip/hip_runtime.h>
#include <hip/hip_bf16.h>

// ---------------------------------------------------------------------------
// LNCC multi-scale loss on (2,1,160,160,160) f32 volumes.
// Separable dilated box-sums expressed as fp32 band-matrix WMMAs
// (v_wmma_f32_16x16x4_f32), then fused LNCC + reduction.
//
// Scale config for 160^3 (from reference): k in {10,20,40}, stride {2,5,10},
// dilation 2 everywhere, weights {0.1,0.3,0.6}.  All shapes are compile-time
// template constants so quantity/row offsets fold into instruction IOFFSETs,
// the band test is branchless (EXEC all-1s around every WMMA per the ISA),
// and the T-loop needs no clamping (T=160 is a multiple of 4).
// ---------------------------------------------------------------------------

typedef float v2f __attribute__((ext_vector_type(2)));
typedef float v8f __attribute__((ext_vector_type(8)));

__device__ __forceinline__ v8f wmma_f32(v2f a, v2f b, v8f c) {
  // 8 args: (neg_a, A, neg_b, B, c_mod, C, reuse_a, reuse_b)
  return __builtin_amdgcn_wmma_f32_16x16x4_f32(false, a, false, b, (short)0, c,
                                               false, false);
}

// Band entry for dilation 2: S[t,o]=1 iff r=t-o*S satisfies 0<=r<2K and r even.
template <int K2>
__device__ __forceinline__ float band1(int r, bool colvalid) {
  return (colvalid && ((unsigned)r < (unsigned)K2) && ((r & 1) == 0)) ? 1.0f
                                                                      : 0.0f;
}

// ---------------------------------------------------------------------------
// Pass W: reduce the contiguous axis.  x,y are [R, T] (R % 16 == 0, T % 4 == 0);
// computes x, y, x^2, y^2, x*y window sums -> dst = 5 arrays of [R, OW].
// One wave computes one 16(rows) x 16(ow) tile for all 5 quantities.
// ---------------------------------------------------------------------------
template <int S, int K, int R, int T, int OW>
__global__ void lncc_pass_w(const float* __restrict__ x,
                            const float* __restrict__ y,
                            float* __restrict__ dst, int tiles) {
  constexpr int owTiles = (OW + 15) / 16;
  int wave = blockIdx.x * (blockDim.x >> 5) + (threadIdx.x >> 5);
  if (wave >= tiles) return;  // wave-uniform exit
  int lane = threadIdx.x & 31;
  int half = lane >> 4;
  int m = lane & 15;

  int tile_r = wave / owTiles;
  int tile_o = wave % owTiles;
  int r0 = tile_r * 16;
  int ow0 = tile_o * 16;

  int row = r0 + m;   // R multiple of 16: always in range
  int owc = ow0 + m;  // B/D-matrix column for this lane
  bool colvalid = owc < OW;
  int ow_s = owc * S;

  int tmin = (ow0 * S) & ~3;
  int tmax = min((ow0 + 15) * S + (K - 1) * 2, T - 1);

  const float* xp = x + (size_t)row * T + tmin + half * 2;
  const float* yp = y + (size_t)row * T + tmin + half * 2;

  v8f acc[5] = {};
  for (int t0 = tmin; t0 <= tmax; t0 += 4, xp += 4, yp += 4) {
    int rr = (t0 + half * 2) - ow_s;
    v2f ax = *(const v2f*)xp;  // 8B-aligned pair -> global_load_b64
    v2f ay = *(const v2f*)yp;
    v2f b = {band1<2 * K>(rr, colvalid), band1<2 * K>(rr + 1, colvalid)};
    v2f ax2 = ax * ax;
    v2f ay2 = ay * ay;
    v2f axy = ax * ay;
    acc[0] = wmma_f32(ax,  b, acc[0]);
    acc[1] = wmma_f32(ay,  b, acc[1]);
    acc[2] = wmma_f32(ax2, b, acc[2]);
    acc[3] = wmma_f32(ay2, b, acc[3]);
    acc[4] = wmma_f32(axy, b, acc[4]);
  }

  constexpr size_t Q = (size_t)R * OW;  // quantity stride (compile-time)
  // Base of this lane's output column; row (r) and quantity (q) offsets are
  // compile-time and fold into instruction offsets where they fit.
  float* dp = dst + (size_t)(r0 + half * 8) * OW + owc;
  if (colvalid) {
    #pragma unroll
    for (int q = 0; q < 5; ++q) {
      #pragma unroll
      for (int r = 0; r < 8; ++r) dp[q * Q + (size_t)r * OW] = acc[q][r];
    }
  }
}

// ---------------------------------------------------------------------------
// Strided-axis pass: src = 5 arrays of [R, T, I] (T % 4 == 0); reduce T -> OT,
// dst = 5 arrays of [R, OT, I].  M-dim over I (stride-1 across lanes), K-dim
// over t (stride I), N-dim over ot.
// ---------------------------------------------------------------------------
template <int S, int K, int R, int T, int I, int OT>
__global__ void lncc_pass_axis(const float* __restrict__ src,
                               float* __restrict__ dst, int tiles) {
  constexpr int iTiles = (I + 15) / 16;
  constexpr int otTiles = (OT + 15) / 16;
  constexpr int tilesPerR = iTiles * otTiles;
  constexpr size_t srcQ = (size_t)R * T * I;
  constexpr size_t dstQ = (size_t)R * OT * I;

  int wave = blockIdx.x * (blockDim.x >> 5) + (threadIdx.x >> 5);
  if (wave >= tiles) return;  // wave-uniform exit
  int lane = threadIdx.x & 31;
  int half = lane >> 4;
  int m = lane & 15;

  int r = wave / tilesPerR;
  int rem = wave % tilesPerR;
  int i0 = (rem / otTiles) * 16;
  int ot0 = (rem % otTiles) * 16;

  int icol = min(i0 + m, I - 1);
  int otc = ot0 + m;
  bool colvalid = otc < OT;
  int ot_s = otc * S;

  int tmin = (ot0 * S) & ~3;
  int tmax = min((ot0 + 15) * S + (K - 1) * 2, T - 1);

  const float* base =
      src + (size_t)r * T * I + icol + (size_t)(tmin + half * 2) * I;

  v8f acc[5] = {};
  for (int t0 = tmin; t0 <= tmax; t0 += 4, base += 4 * (size_t)I) {
    int rr = (t0 + half * 2) - ot_s;
    v2f b = {band1<2 * K>(rr, colvalid), band1<2 * K>(rr + 1, colvalid)};
    #pragma unroll
    for (int q = 0; q < 5; ++q) {
      // q*srcQ and +I are compile-time: immediate-offset loads off one base.
      v2f a = {base[q * srcQ], base[q * srcQ + I]};
      acc[q] = wmma_f32(a, b, acc[q]);
    }
  }

  // Output tile: element (M=iout, N=otc); M runs over the stride-1 axis, so
  // the 8 per-quantity stores are at consecutive +4B compile-time offsets.
  float* dp = dst + (size_t)r * OT * I + (size_t)otc * I + (i0 + half * 8);
  #pragma unroll
  for (int q = 0; q < 5; ++q) {
    #pragma unroll
    for (int rr = 0; rr < 8; ++rr) {
      int iout = i0 + half * 8 + rr;
      if (iout < I && colvalid) dp[q * dstQ + rr] = acc[q][rr];
    }
  }
}

// ---------------------------------------------------------------------------
// Final: per-window LNCC from the 5 fully-reduced arrays, block reduction,
// atomic accumulate of  -w/N * sum(lncc)  into out[0].
// ---------------------------------------------------------------------------
__global__ void lncc_final(const float* __restrict__ buf, size_t N, float numel,
                           float coef, float* __restrict__ out) {
  __shared__ float red[256];
  size_t i = (size_t)blockIdx.x * blockDim.x + threadIdx.x;
  size_t stride = (size_t)gridDim.x * blockDim.x;
  float sum = 0.0f;
  for (; i < N; i += stride) {
    float xs = buf[i];
    float ys = buf[N + i];
    float x2s = buf[2 * N + i];
    float y2s = buf[3 * N + i];
    float xys = buf[4 * N + i];
    float xm = xs / numel;
    float ym = ys / numel;
    float cross = xys - ym * xs - xm * ys + ym * xm * numel;
    float xvar = x2s - 2.0f * xm * xs + xm * xm * numel;
    float yvar = y2s - 2.0f * ym * ys + ym * ym * numel;
    sum += cross * cross / (xvar * yvar + 1e-5f);
  }
  red[threadIdx.x] = sum;
  __syncthreads();
  for (int off = 128; off > 0; off >>= 1) {
    if ((int)threadIdx.x < off) red[threadIdx.x] += red[threadIdx.x + off];
    __syncthreads();
  }
  if (threadIdx.x == 0) atomicAdd(out, coef * red[0]);
}

__global__ void lncc_init(float* __restrict__ out) {
  if (threadIdx.x == 0) out[0] = 1.0f;  // sum of scale weights 0.1+0.3+0.6
}

// ---------------------------------------------------------------------------
template <int S, int K>
static void run_scale(const float* x, const float* y, float* ws, float* out,
                      float w, hipStream_t stream) {
  constexpr int B = 2, D = 160, H = 160, W = 160;
  constexpr int span = (K - 1) * 2 + 1;
  constexpr int OW = (W - span) / S + 1;  // 71 / 25 / 9
  constexpr size_t OFF_B = 18176000;      // floats; pass-W region max (k=10)

  float* bufA = ws;
  float* bufB = ws + OFF_B;
  float* bufC = ws;  // pass-Z output aliases bufA (reads only bufB)

  // Pass W: [B*D*H, W] -> 5 x [B*D*H, OW]
  {
    constexpr int R = B * D * H;  // 51200, multiple of 16
    constexpr int owTiles = (OW + 15) / 16;
    constexpr int tiles = (R / 16) * owTiles;
    lncc_pass_w<S, K, R, W, OW>
        <<<(tiles + 3) / 4, 128, 0, stream>>>(x, y, bufA, tiles);
  }
  // Pass H: 5 x [B*D, H, OW] -> 5 x [B*D, OW, OW]
  {
    constexpr int R = B * D;
    constexpr int iTiles = (OW + 15) / 16, otTiles = (OW + 15) / 16;
    constexpr int tiles = R * iTiles * otTiles;
    lncc_pass_axis<S, K, R, H, OW, OW>
        <<<(tiles + 3) / 4, 128, 0, stream>>>(bufA, bufB, tiles);
  }
  // Pass Z: 5 x [B, D, OW*OW] -> 5 x [B, OW, OW*OW]
  {
    constexpr int R = B, I = OW * OW;
    constexpr int iTiles = (I + 15) / 16, otTiles = (OW + 15) / 16;
    constexpr int tiles = R * iTiles * otTiles;
    lncc_pass_axis<S, K, R, D, I, OW>
        <<<(tiles + 3) / 4, 128, 0, stream>>>(bufB, bufC, tiles);
  }
  // Final LNCC + weighted-mean accumulation
  {
    constexpr size_t N = (size_t)B * OW * OW * OW;
    constexpr float numel = (float)K * (float)K * (float)K;
    constexpr size_t want = (N + 255) / 256;
    constexpr int blocks = (int)(want > 2048 ? 2048 : (want < 1 ? 1 : want));
    lncc_final<<<blocks, 256, 0, stream>>>(bufC, N, numel, -w / (float)N, out);
  }
}

extern "C" void kernel_launch(void* const* d_in, const int* in_sizes, int n_in,
                              void* d_out, int out_size, void* d_ws,
                              size_t ws_size, hipStream_t stream) {
  const float* x = (const float*)d_in[0];
  const float* y = (const float*)d_in[1];
  float* out = (float*)d_out;
  float* ws = (float*)d_ws;

  lncc_init<<<1, 32, 0, stream>>>(out);

  run_scale<2, 10>(x, y, ws, out, 0.1f, stream);
  run_scale<5, 20>(x, y, ws, out, 0.3f, stream);
  run_scale<10, 40>(x, y, ws, out, 0.6f, stream);
}